// GAT_pyg_59313498357783
// MI455X (gfx1250) — compile-verified
//
#include <hip/hip_runtime.h>
#include <math.h>

#define D 32

typedef float v2f __attribute__((ext_vector_type(2)));
typedef float v8f __attribute__((ext_vector_type(8)));

// ---------------------------------------------------------------------------
// float atomic max via sign-aware integer atomics (works with -inf init)
// ---------------------------------------------------------------------------
__device__ __forceinline__ void atomic_max_f32(float* addr, float val) {
    if (val >= 0.0f) {
        atomicMax((int*)addr, __float_as_int(val));
    } else {
        atomicMin((unsigned int*)addr, __float_as_uint(val));
    }
}

__device__ __forceinline__ void edge_nodes(const int* __restrict__ ei, int E,
                                           int e, int& s, int& d) {
    if (e < E) { s = ei[e]; d = ei[E + e]; }
    else       { s = e - E; d = e - E; }     // self loops appended
}

// ---------------------------------------------------------------------------
// h = X @ W   (X: n x 32 fp32, W: 32 x 32 fp32)  via V_WMMA_F32_16X16X4_F32
// One wave32 computes a 16-row tile (two 16-col accumulators), K=32 in 8x K=4.
// ---------------------------------------------------------------------------
__global__ void gat_linear_wmma(const float* __restrict__ X,
                                const float* __restrict__ W,
                                float* __restrict__ H, int n_nodes) {
    __shared__ float sW[D * D];
    for (int i = threadIdx.x; i < D * D; i += blockDim.x) sW[i] = W[i];
    __syncthreads();

    const int wave = threadIdx.x >> 5;
    const int lane = threadIdx.x & 31;
    const int tile = blockIdx.x * (blockDim.x >> 5) + wave;
    const int row0 = tile * 16;
    if (row0 >= n_nodes) return;                 // wave-uniform exit

    const int m  = lane & 15;                    // M (A) / N (B,C,D) index
    const int kh = (lane >> 4) << 1;             // K sub-offset: 0 or 2

    int arow = row0 + m;
    if (arow >= n_nodes) arow = n_nodes - 1;     // clamp: keep EXEC full
    const float* xrow = X + (size_t)arow * D;

    v8f c0 = {}; v8f c1 = {};
#pragma unroll
    for (int k0 = 0; k0 < D; k0 += 4) {
        // A 16x4 fp32: lane m, VGPR0 = K=(k0+kh), VGPR1 = K=(k0+kh+1)
        const float2 av = *(const float2*)(xrow + k0 + kh);
        v2f a; a.x = av.x; a.y = av.y;
        // B 4x16 fp32: lane n=m (cols), same K striping as A
        v2f b0, b1;
        b0.x = sW[(k0 + kh) * D + m];
        b0.y = sW[(k0 + kh + 1) * D + m];
        b1.x = sW[(k0 + kh) * D + m + 16];
        b1.y = sW[(k0 + kh + 1) * D + m + 16];
        c0 = __builtin_amdgcn_wmma_f32_16x16x4_f32(false, a, false, b0,
                                                   (short)0, c0, false, false);
        c1 = __builtin_amdgcn_wmma_f32_16x16x4_f32(false, a, false, b1,
                                                   (short)0, c1, false, false);
    }

    // C/D 16x16 fp32: VGPR r -> M=r (lanes 0-15), M=r+8 (lanes 16-31); N=lane&15
    const int rbase = row0 + ((lane >> 4) << 3);
#pragma unroll
    for (int r = 0; r < 8; ++r) {
        const int row = rbase + r;
        if (row < n_nodes) {
            H[(size_t)row * D + m]      = c0[r];
            H[(size_t)row * D + 16 + m] = c1[r];
        }
    }
}

// ---------------------------------------------------------------------------
// alpha_src[i] = h[i,:].a_src ; alpha_dst[i] = h[i,:].a_dst
// ---------------------------------------------------------------------------
__global__ void gat_alpha(const float* __restrict__ H,
                          const float* __restrict__ a_src,
                          const float* __restrict__ a_dst,
                          float* __restrict__ asrc,
                          float* __restrict__ adst, int n) {
    int i = blockIdx.x * blockDim.x + threadIdx.x;
    if (i >= n) return;
    const float4* hr = (const float4*)(H + (size_t)i * D);
    const float4* s4 = (const float4*)a_src;
    const float4* d4 = (const float4*)a_dst;
    float ss = 0.f, dd = 0.f;
#pragma unroll
    for (int q = 0; q < D / 4; ++q) {
        float4 h = hr[q], s = s4[q], d = d4[q];
        ss += h.x * s.x + h.y * s.y + h.z * s.z + h.w * s.w;
        dd += h.x * d.x + h.y * d.y + h.z * d.z + h.w * d.w;
    }
    asrc[i] = ss; adst[i] = dd;
}

// ---------------------------------------------------------------------------
// init: agg <- bias (broadcast), mbuf <- -inf, sbuf <- 0
// ---------------------------------------------------------------------------
__global__ void gat_init(float* __restrict__ agg, const float* __restrict__ bias,
                         float* __restrict__ mbuf, float* __restrict__ sbuf,
                         int n) {
    int i = blockIdx.x * blockDim.x + threadIdx.x;
    if (i < n * D) agg[i] = bias[i & (D - 1)];
    if (i < n) { mbuf[i] = -INFINITY; sbuf[i] = 0.0f; }
}

// ---------------------------------------------------------------------------
// pass 1: segment max of leaky_relu logits over destination
// ---------------------------------------------------------------------------
__global__ void gat_edge_max(const int* __restrict__ ei, int E, int Etot,
                             const float* __restrict__ asrc,
                             const float* __restrict__ adst,
                             float* __restrict__ mbuf) {
    int e = blockIdx.x * blockDim.x + threadIdx.x;
    if (e >= Etot) return;
    int s, d; edge_nodes(ei, E, e, s, d);
    float a = asrc[s] + adst[d];
    a = a > 0.0f ? a : 0.2f * a;
    atomic_max_f32(&mbuf[d], a);
}

// ---------------------------------------------------------------------------
// pass 2: segment sum of exp(logit - max)
// ---------------------------------------------------------------------------
__global__ void gat_edge_sum(const int* __restrict__ ei, int E, int Etot,
                             const float* __restrict__ asrc,
                             const float* __restrict__ adst,
                             const float* __restrict__ mbuf,
                             float* __restrict__ sbuf) {
    int e = blockIdx.x * blockDim.x + threadIdx.x;
    if (e >= Etot) return;
    int s, d; edge_nodes(ei, E, e, s, d);
    float a = asrc[s] + adst[d];
    a = a > 0.0f ? a : 0.2f * a;
    atomicAdd(&sbuf[d], __expf(a - mbuf[d]));
}

// ---------------------------------------------------------------------------
// pass 3: one wave32 per edge; lane j scatters feature j (coalesced 128B)
// ---------------------------------------------------------------------------
__global__ void gat_edge_scatter(const int* __restrict__ ei, int E, int Etot,
                                 const float* __restrict__ asrc,
                                 const float* __restrict__ adst,
                                 const float* __restrict__ mbuf,
                                 const float* __restrict__ sbuf,
                                 const float* __restrict__ H,
                                 float* __restrict__ agg) {
    int e = blockIdx.x * (blockDim.x >> 5) + (threadIdx.x >> 5);
    if (e >= Etot) return;
    int lane = threadIdx.x & 31;
    int s, d; edge_nodes(ei, E, e, s, d);
    float a = asrc[s] + adst[d];                 // broadcast loads, L2-resident
    a = a > 0.0f ? a : 0.2f * a;
    float coef = __expf(a - mbuf[d]) / fmaxf(sbuf[d], 1e-16f);
    atomicAdd(&agg[(size_t)d * D + lane], H[(size_t)s * D + lane] * coef);
}

// ---------------------------------------------------------------------------
// pass 4: in-place ELU
// ---------------------------------------------------------------------------
__global__ void gat_elu(float* __restrict__ a, int total) {
    int i = blockIdx.x * blockDim.x + threadIdx.x;
    if (i >= total) return;
    float x = a[i];
    a[i] = x > 0.0f ? x : (__expf(x) - 1.0f);
}

// ---------------------------------------------------------------------------
extern "C" void kernel_launch(void* const* d_in, const int* in_sizes, int n_in,
                              void* d_out, int out_size, void* d_ws, size_t ws_size,
                              hipStream_t stream) {
    const float* x   = (const float*)d_in[0];
    const int*   ei  = (const int*)  d_in[1];
    const float* W1  = (const float*)d_in[2];
    const float* as1 = (const float*)d_in[3];
    const float* ad1 = (const float*)d_in[4];
    const float* b1  = (const float*)d_in[5];
    const float* W2  = (const float*)d_in[6];
    const float* as2 = (const float*)d_in[7];
    const float* ad2 = (const float*)d_in[8];
    const float* b2  = (const float*)d_in[9];

    const int n    = in_sizes[0] / D;
    const int E    = in_sizes[1] / 2;
    const int Etot = E + n;              // self-loops appended

    float* out  = (float*)d_out;
    float* xbar = out;                   // first output  [n*D]
    float* z    = out + (size_t)n * D;   // second output [n*D]

    float* H    = (float*)d_ws;          // n*D
    float* asrc = H + (size_t)n * D;     // n
    float* adst = asrc + n;              // n
    float* mbuf = adst + n;              // n
    float* sbuf = mbuf + n;              // n

    const int BT = 256;                  // 8 wave32s per block
    dim3 blk(BT);
    const int wavesPerBlock = BT / 32;
    const int nTiles = (n + 15) / 16;
    dim3 gemmGrid((nTiles + wavesPerBlock - 1) / wavesPerBlock);
    dim3 nodeGrid((n + BT - 1) / BT);
    dim3 ndGrid((n * D + BT - 1) / BT);
    dim3 edgeGrid((Etot + BT - 1) / BT);
    dim3 scatGrid((Etot + wavesPerBlock - 1) / wavesPerBlock);

    // ----- layer 1: h1 = x@W1 ; z = elu(gat_aggregate(h1) + b1)
    gat_linear_wmma<<<gemmGrid, blk, 0, stream>>>(x, W1, H, n);
    gat_alpha      <<<nodeGrid, blk, 0, stream>>>(H, as1, ad1, asrc, adst, n);
    gat_init       <<<ndGrid,   blk, 0, stream>>>(z, b1, mbuf, sbuf, n);
    gat_edge_max   <<<edgeGrid, blk, 0, stream>>>(ei, E, Etot, asrc, adst, mbuf);
    gat_edge_sum   <<<edgeGrid, blk, 0, stream>>>(ei, E, Etot, asrc, adst, mbuf, sbuf);
    gat_edge_scatter<<<scatGrid, blk, 0, stream>>>(ei, E, Etot, asrc, adst, mbuf, sbuf, H, z);
    gat_elu        <<<ndGrid,   blk, 0, stream>>>(z, n * D);

    // ----- layer 2: h2 = z@W2 ; xbar = elu(gat_aggregate(h2) + b2)
    gat_linear_wmma<<<gemmGrid, blk, 0, stream>>>(z, W2, H, n);
    gat_alpha      <<<nodeGrid, blk, 0, stream>>>(H, as2, ad2, asrc, adst, n);
    gat_init       <<<ndGrid,   blk, 0, stream>>>(xbar, b2, mbuf, sbuf, n);
    gat_edge_max   <<<edgeGrid, blk, 0, stream>>>(ei, E, Etot, asrc, adst, mbuf);
    gat_edge_sum   <<<edgeGrid, blk, 0, stream>>>(ei, E, Etot, asrc, adst, mbuf, sbuf);
    gat_edge_scatter<<<scatGrid, blk, 0, stream>>>(ei, E, Etot, asrc, adst, mbuf, sbuf, H, xbar);
    gat_elu        <<<ndGrid,   blk, 0, stream>>>(xbar, n * D);
}